// HiggsAudioRVQ_88656714924736
// MI455X (gfx1250) — compile-verified
//
#include <hip/hip_runtime.h>
#include <hip/hip_bf16.h>

// Problem constants (from reference setup_inputs):
//   Q=8, B=8, T=8192, K=1024, D=64, H=1024
#define Qn 8
#define Bn 8
#define Tn 8192
#define Kn 1024
#define Dn 64
#define Hn 1024

typedef __attribute__((ext_vector_type(2))) float v2f;
typedef __attribute__((ext_vector_type(8))) float v8f;

// -----------------------------------------------------------------------------
// Kernel 1: P[q,k,h] = sum_d CB[q,k,d] * W[q,h,d] + bias[q,h]
// One wave32 computes one 16x16 (k x h) tile of P for one q via
// V_WMMA_F32_16X16X4_F32, looping d in steps of 4 (16 WMMA ops per wave).
//
// 32-bit A-matrix 16x4 layout: lane<16 holds row M=lane, K={0,1} in v0/v1;
// lane>=16 holds row M=lane-16, K={2,3}.  B (=W^T, 4x16) is symmetric with
// N=lane&15.  C/D: VGPR r holds M = r + 8*(lane>>4), N = lane&15.
// -----------------------------------------------------------------------------
__global__ __launch_bounds__(256) void rvq_project_wmma(
    const float* __restrict__ CB,   // [Q,K,D]
    const float* __restrict__ W,    // [Q,H,D]
    const float* __restrict__ bias, // [Q,H]
    float* __restrict__ P)          // [Q,K,H]  (workspace)
{
    const int lane = threadIdx.x & 31;
    const int wave = threadIdx.x >> 5;
    const int wg   = blockIdx.x * 8 + wave;      // 0 .. Q*(K/16)*(H/16)-1 = 32767
    const int q    = wg >> 12;                   // / (64*64)
    const int rem  = wg & 4095;
    const int k0   = (rem >> 6) << 4;            // k tile base
    const int h0   = (rem & 63) << 4;            // h tile base

    const int mrow = lane & 15;                  // M (or N) row within tile
    const int koff = (lane >> 4) << 1;           // K sub-offset: 0 or 2

    const float* pa = CB + (size_t)(q * Kn + k0 + mrow) * Dn + koff; // A = CB
    const float* pb = W  + (size_t)(q * Hn + h0 + mrow) * Dn + koff; // B = W^T

    v8f acc = {};
#pragma unroll
    for (int d0 = 0; d0 < Dn; d0 += 4) {
        v2f afrag = *(const v2f*)(pa + d0);      // CB[k0+m][d0+koff .. +1]
        v2f bfrag = *(const v2f*)(pb + d0);      // W [h0+n][d0+koff .. +1]
        acc = __builtin_amdgcn_wmma_f32_16x16x4_f32(
            /*neg_a=*/false, afrag, /*neg_b=*/false, bfrag,
            /*c_mod=*/(short)0, acc, /*reuse_a=*/false, /*reuse_b=*/false);
    }

    // Epilogue: fold bias[q,h] into every row (so kernel 2 never touches bias).
    const int nrow = lane & 15;
    const int mb   = (lane >> 4) << 3;           // 0 or 8
    const float bv = bias[q * Hn + h0 + nrow];
#pragma unroll
    for (int r = 0; r < 8; ++r) {
        P[((size_t)(q * Kn + k0 + mb + r) << 10) + (h0 + nrow)] = acc[r] + bv;
    }
}

// -----------------------------------------------------------------------------
// Kernel 2: out[b,h,t] = sum_q P[q, codes[q,b,t], h]
// 64h x 64t tile per 256-thread workgroup.  Accumulate phase reads P with
// lanes contiguous in h (coalesced 128B L2 hits; P is 32MB, L2-resident).
// LDS transpose (64x65 padded -> conflict-free), then coalesced stores along t.
// -----------------------------------------------------------------------------
__global__ __launch_bounds__(256) void rvq_gather_sum(
    const int*   __restrict__ codes, // [Q,B,T] (int32 per harness contract)
    const float* __restrict__ P,     // [Q,K,H]
    float*       __restrict__ out)   // [B,H,T]
{
    __shared__ int   scode[Qn][64];
    __shared__ float tile[64][65];

    const int b   = blockIdx.z;
    const int h0  = blockIdx.y << 6;
    const int t0  = blockIdx.x << 6;
    const int tid = threadIdx.x;

    // Stage the 8*64 code indices for this (b, t-tile).
#pragma unroll
    for (int i = tid; i < Qn * 64; i += 256) {
        const int q = i >> 6, tt = i & 63;
        scode[q][tt] = codes[((size_t)(q * Bn + b) << 13) + t0 + tt];
    }
    __syncthreads();

    // Accumulate: thread -> (h_local = tid&63, t strided by 4).
    // Lanes 0..31 of each wave have consecutive h -> 128B coalesced P reads.
    const int hl = tid & 63;
    const int tg = tid >> 6;
    for (int tt = tg; tt < 64; tt += 4) {
        float acc = 0.0f;
#pragma unroll
        for (int q = 0; q < Qn; ++q) {
            const int c = scode[q][tt];
            acc += P[((size_t)((q << 10) + c) << 10) + h0 + hl];
        }
        tile[hl][tt] = acc;   // stride-65 column write: conflict-free
    }
    __syncthreads();

    // Write: thread -> (t_local = tid&63, h strided by 4).
    // Lanes have consecutive t -> 128B coalesced HBM stores into [B,H,T].
    const int tl = tid & 63;
    const int hg = tid >> 6;
    for (int hh = hg; hh < 64; hh += 4) {
        out[((size_t)(b * Hn + h0 + hh) << 13) + t0 + tl] = tile[hh][tl];
    }
}

extern "C" void kernel_launch(void* const* d_in, const int* in_sizes, int n_in,
                              void* d_out, int out_size, void* d_ws, size_t ws_size,
                              hipStream_t stream) {
    const int*   codes = (const int*)  d_in[0];  // [Q,B,T]
    const float* CB    = (const float*)d_in[1];  // [Q,K,D]
    const float* W     = (const float*)d_in[2];  // [Q,H,D]
    const float* bias  = (const float*)d_in[3];  // [Q,H]
    float*       out   = (float*)d_out;          // [B,H,T]
    float*       P     = (float*)d_ws;           // [Q,K,H] = 32 MB scratch

    // Phase 1: projected-codebook GEMM (WMMA), bias folded in.
    // Q*(K/16)*(H/16) = 32768 wave-tiles, 8 waves per 256-thread block.
    rvq_project_wmma<<<dim3(4096), dim3(256), 0, stream>>>(CB, W, bias, P);

    // Phase 2: L2-resident gather + sum, coalesced transpose-store.
    dim3 grid2(Tn / 64, Hn / 64, Bn);            // 128 x 16 x 8 = 16384 blocks
    rvq_gather_sum<<<grid2, dim3(256), 0, stream>>>(codes, P, out);
}